// DoubleSoftmax_32298154066218
// MI455X (gfx1250) — compile-verified
//
#include <hip/hip_runtime.h>
#include <hip/hip_bf16.h>

// Problem constants (from reference): B=2, N=768, F=H=128, K_IN=2, K=2
#define Bb     2
#define Nn     768
#define Ff     128
#define Hh     128
#define Dd     64                     // H / K_IN
#define TILES  (Nn / 16)              // 48 16-row tiles per dim
#define PAIRS_PER_B (TILES * (TILES + 1) / 2)   // 1176 upper-tri tile pairs
#define TOTAL_PAIRS (Bb * PAIRS_PER_B)          // 2352 (divisible by 8)

typedef __attribute__((ext_vector_type(2))) float v2f;
typedef __attribute__((ext_vector_type(8))) float v8f;

// ---------------------------------------------------------------------------
// Kernel 1: Xc[b,k,n,d] = X[b,n,2d+k] + gate * feats[b,n,2d+k]
// feats = relu(X @ ef_w1 + b1) @ ef_w2 + b2.  One row (b,n) per 128-thread
// block; weight-column reads are lane-coalesced; X row / hidden row staged in
// LDS.  ~0.1 GFLOP total — negligible.
// ---------------------------------------------------------------------------
__global__ __launch_bounds__(128) void xc_kernel(
    const float* __restrict__ X,
    const float* __restrict__ ef_w1, const float* __restrict__ ef_b1,
    const float* __restrict__ ef_w2, const float* __restrict__ ef_b2,
    const float* __restrict__ gate,
    float* __restrict__ Xc)
{
    __shared__ float xrow[Ff];
    __shared__ float hrow[Hh];
    const int t   = threadIdx.x;
    const int row = blockIdx.x;          // row = b*Nn + n
    const int b   = row / Nn;
    const int n   = row - b * Nn;

    xrow[t] = X[(size_t)row * Ff + t];
    __syncthreads();

    float a = ef_b1[t];
    #pragma unroll 4
    for (int f = 0; f < Ff; ++f)
        a = fmaf(xrow[f], ef_w1[f * Hh + t], a);   // coalesced across lanes
    hrow[t] = fmaxf(a, 0.0f);
    __syncthreads();

    float a2 = ef_b2[t];
    #pragma unroll 4
    for (int h = 0; h < Hh; ++h)
        a2 = fmaf(hrow[h], ef_w2[h * Hh + t], a2);

    const float xc = xrow[t] + gate[0] * a2;
    const int d = t >> 1;                // feature f = 2d + k
    const int k = t & 1;
    Xc[(((size_t)(b * 2 + k)) * Nn + n) * Dd + d] = xc;
}

// ---------------------------------------------------------------------------
// Kernel 2: fused  prod (WMMA f32 16x16x4)  ->  edge MLP (VALU + LDS table)
//           ->  gumbel sign test  ->  triu + symmetric store.
// One wave per upper-triangular 16x16 (i,j) tile.  prod never touches HBM.
// ---------------------------------------------------------------------------
__global__ __launch_bounds__(256) void edge_kernel(
    const float* __restrict__ Xc,
    const float* __restrict__ sm_w1, const float* __restrict__ sm_b1,
    const float* __restrict__ sm_w2, const float* __restrict__ sm_b2,
    const float* __restrict__ gumbel,
    float* __restrict__ out)
{
    // Packed per-h MLP table: {w1[0,h], w1[1,h], b1[h], w2[h,1]-w2[h,0]}
    __shared__ float4 mlp[Hh];
    const int t = threadIdx.x;
    if (t < Hh)
        mlp[t] = make_float4(sm_w1[t], sm_w1[Hh + t], sm_b1[t],
                             sm_w2[2 * t + 1] - sm_w2[2 * t]);
    __syncthreads();

    const int wave = t >> 5;
    const int lane = t & 31;
    const int pair = blockIdx.x * 8 + wave;       // wave-uniform
    if (pair >= TOTAL_PAIRS) return;

    // Decode (b, ti, tj) with tj >= ti from linear upper-tri index.
    const int b = pair / PAIRS_PER_B;
    int q = pair - b * PAIRS_PER_B;
    int ti = 0;
    while (q >= TILES - ti) { q -= TILES - ti; ++ti; }
    const int tj = ti + q;
    const int i0 = ti * 16, j0 = tj * 16;

    const int r  = lane & 15;     // N-index within tile / A,B row select
    const int hi = lane >> 4;     // selects K pair {0,1} vs {2,3} and M+8 half

    // --- pairwise product via V_WMMA_F32_16X16X4_F32 -----------------------
    // A[i,Kd] = Xc[b,k,i0+i, d0+Kd]; B[Kd,j] = Xc[b,k,j0+j, d0+Kd].
    // Both fragments are the same 8B/lane gather (row r, cols d0 + 2*hi).
    v8f p0 = {}, p1 = {};
    {
        const float* bi = Xc + ((size_t)(b * 2 + 0) * Nn + i0 + r) * Dd + 2 * hi;
        const float* bj = Xc + ((size_t)(b * 2 + 0) * Nn + j0 + r) * Dd + 2 * hi;
        #pragma unroll
        for (int s = 0; s < 16; ++s) {
            v2f av = *(const v2f*)(bi + 4 * s);
            v2f bv = *(const v2f*)(bj + 4 * s);
            p0 = __builtin_amdgcn_wmma_f32_16x16x4_f32(
                     false, av, false, bv, (short)0, p0, false, false);
        }
    }
    {
        const float* bi = Xc + ((size_t)(b * 2 + 1) * Nn + i0 + r) * Dd + 2 * hi;
        const float* bj = Xc + ((size_t)(b * 2 + 1) * Nn + j0 + r) * Dd + 2 * hi;
        #pragma unroll
        for (int s = 0; s < 16; ++s) {
            v2f av = *(const v2f*)(bi + 4 * s);
            v2f bv = *(const v2f*)(bj + 4 * s);
            p1 = __builtin_amdgcn_wmma_f32_16x16x4_f32(
                     false, av, false, bv, (short)0, p1, false, false);
        }
    }

    // --- per-edge score difference:  s1 - s0 -------------------------------
    // C-layout: acc element e -> (i = i0 + e + 8*hi, j = j0 + r)
    const float b2d = sm_b2[1] - sm_b2[0];
    float acc[8];
    #pragma unroll
    for (int e = 0; e < 8; ++e) acc[e] = b2d;

    #pragma unroll 4
    for (int h = 0; h < Hh; ++h) {
        const float4 m = mlp[h];            // LDS broadcast, conflict-free
        #pragma unroll
        for (int e = 0; e < 8; ++e) {
            float pre = fmaf(p0[e], m.x, fmaf(p1[e], m.y, m.z));
            acc[e] = fmaf(fmaxf(pre, 0.0f), m.w, acc[e]);
        }
    }

    // --- gumbel argmax sign test + triu + symmetrize -----------------------
    const int j = j0 + r;
    #pragma unroll
    for (int e = 0; e < 8; ++e) {
        const int i = i0 + e + 8 * hi;
        if (j > i) {
            const float* gp = gumbel + (((size_t)b * Nn + i) * Nn + j) * 2;
            const float g = gp[1] - gp[0];
            const float dec = (acc[e] + g) > 0.0f ? 1.0f : 0.0f;
            out[((size_t)b * Nn + i) * Nn + j] = dec;
            out[((size_t)b * Nn + j) * Nn + i] = dec;   // mirror
        } else if (j == i) {
            out[((size_t)b * Nn + i) * Nn + j] = 0.0f;  // zero diagonal
        }
    }
}

// ---------------------------------------------------------------------------
extern "C" void kernel_launch(void* const* d_in, const int* in_sizes, int n_in,
                              void* d_out, int out_size, void* d_ws, size_t ws_size,
                              hipStream_t stream)
{
    const float* X      = (const float*)d_in[0];
    const float* ef_w1  = (const float*)d_in[1];
    const float* ef_b1  = (const float*)d_in[2];
    const float* ef_w2  = (const float*)d_in[3];
    const float* ef_b2  = (const float*)d_in[4];
    const float* gate   = (const float*)d_in[5];
    const float* sm_w1  = (const float*)d_in[6];
    const float* sm_b1  = (const float*)d_in[7];
    const float* sm_w2  = (const float*)d_in[8];
    const float* sm_b2  = (const float*)d_in[9];
    const float* gumbel = (const float*)d_in[10];
    float* out = (float*)d_out;

    float* Xc = (float*)d_ws;   // [B,2,N,64] fp32 = 786 KB (fits L2 easily)

    xc_kernel<<<Bb * Nn, 128, 0, stream>>>(X, ef_w1, ef_b1, ef_w2, ef_b2, gate, Xc);
    edge_kernel<<<TOTAL_PAIRS / 8, 256, 0, stream>>>(Xc, sm_w1, sm_b1, sm_w2,
                                                     sm_b2, gumbel, out);
}